// GNNResGatedGraphConv_58136677319332
// MI455X (gfx1250) — compile-verified
//
#include <hip/hip_runtime.h>

typedef __attribute__((ext_vector_type(2))) float v2f;
typedef __attribute__((ext_vector_type(8))) float v8f;

constexpr int   NN   = 100000;   // nodes
constexpr int   NE   = 800000;   // edges
constexpr int   DD   = 64;       // node feature dim
constexpr int   EDIM = 8;        // edge feature dim
constexpr int   EPB  = 16;       // edges per block in edge kernel
constexpr float EPS  = 1e-5f;

// ---------------------------------------------------------------------------
// Zero the aggregation buffer and the BN stats accumulator.
// ---------------------------------------------------------------------------
__global__ __launch_bounds__(256) void zero_kernel(float* __restrict__ agg,
                                                   float* __restrict__ stats,
                                                   int n) {
  int i = blockIdx.x * 256 + threadIdx.x;
  if (i < n) agg[i] = 0.0f;
  if (i < 2 * DD) stats[i] = 0.0f;
}

// ---------------------------------------------------------------------------
// Node GEMM via V_WMMA_F32_16X16X4_F32:
//   [K|Q|V|S] = X @ [Wk|Wq|Wv|Ws] + bias
// One 512-thread block (16 waves) per 16-row tile of X. The 16x64 A-tile is
// staged once into LDS and shared by all 16 waves; wave w computes output
// tile (matrix = w>>2, colTile = w&3).
// A-frag layout (32-bit A 16x4, ISA 7.12.2): lanes 0-15 hold K={kb,kb+1},
// lanes 16-31 hold K={kb+2,kb+3}, row M = lane&15.
// C/D layout: VGPR r -> row (lane<16 ? r : r+8), col = lane&15.
// ---------------------------------------------------------------------------
__global__ __launch_bounds__(512) void gemm_node_wmma(
    const float* __restrict__ X,
    const float* __restrict__ Wk, const float* __restrict__ bk,
    const float* __restrict__ Wq, const float* __restrict__ bq,
    const float* __restrict__ Wv, const float* __restrict__ bv,
    const float* __restrict__ Ws, const float* __restrict__ bs,
    float* __restrict__ Ko, float* __restrict__ Qo,
    float* __restrict__ Vo, float* __restrict__ So) {
  __shared__ float sX[16 * DD];                // 4 KB A-tile

  const int tid  = threadIdx.y * 32 + threadIdx.x;
  const int row0 = blockIdx.x * 16;

  // Coalesced stage of the 16x64 X tile (1024 floats, 512 threads).
  for (int i = tid; i < 16 * DD; i += 512)
    sX[i] = X[(size_t)row0 * DD + i];
  __syncthreads();

  const int wave = threadIdx.y;               // 0..15
  const int mat  = wave >> 2;                 // which weight matrix
  const int ct   = wave & 3;                  // 16-col tile within 64
  const int lane = threadIdx.x;               // 0..31 (wave32)
  const int mrow = lane & 15;
  const int kh   = lane >> 4;                 // K-half select
  const int n    = (lane & 15) + ct * 16;     // output column 0..63

  const float* W; const float* B; float* O;
  if      (mat == 0) { W = Wk; B = bk; O = Ko; }
  else if (mat == 1) { W = Wq; B = bq; O = Qo; }
  else if (mat == 2) { W = Wv; B = bv; O = Vo; }
  else               { W = Ws; B = bs; O = So; }

  const float bias = B[n];
  v8f acc;
#pragma unroll
  for (int r = 0; r < 8; ++r) acc[r] = bias;

#pragma unroll
  for (int kb = 0; kb < DD; kb += 4) {
    const int k0 = kb + 2 * kh;
    v2f a, b;
    a.x = sX[mrow * DD + k0];
    a.y = sX[mrow * DD + k0 + 1];
    b.x = W[k0 * DD + n];
    b.y = W[(k0 + 1) * DD + n];
    acc = __builtin_amdgcn_wmma_f32_16x16x4_f32(
        /*neg_a=*/false, a, /*neg_b=*/false, b,
        /*c_mod=*/(short)0, acc, /*reuse_a=*/false, /*reuse_b=*/false);
  }

#pragma unroll
  for (int r = 0; r < 8; ++r) {
    const int row = row0 + (kh ? r + 8 : r);
    O[(size_t)row * DD + n] = acc[r];
  }
}

// ---------------------------------------------------------------------------
// Fused edge stage. 16 edges per 256-thread block; 16 threads per edge, each
// covering 4 contiguous channels. x_edge + src/dst staged in LDS (coalesced);
// ep = x_edge @ We + be computed on the fly; k/q/v gathered with b128 loads;
// gated message scatter-added with global_atomic_add_f32.
// ---------------------------------------------------------------------------
__global__ __launch_bounds__(256) void edge_kernel(
    const int* __restrict__ ei,         // [2, E]: row0 = src, row1 = dst
    const float* __restrict__ XE,       // [E, 8]
    const float* __restrict__ We,       // [8, 64]
    const float* __restrict__ be,       // [64]
    const float* __restrict__ K,
    const float* __restrict__ Q,
    const float* __restrict__ V,
    float* __restrict__ AGG) {
  __shared__ float sWe[EDIM * DD];      // 2 KB
  __shared__ float sbe[DD];
  __shared__ float sXE[EPB * EDIM];     // 128 floats
  __shared__ int   sSrc[EPB];
  __shared__ int   sDst[EPB];

  const int tid   = threadIdx.x;
  const int ebase = blockIdx.x * EPB;

  for (int i = tid; i < EDIM * DD; i += 256) sWe[i] = We[i];
  if (tid < DD) sbe[tid] = be[tid];
  if (tid < EPB * EDIM) sXE[tid] = XE[(size_t)ebase * EDIM + tid];
  if (tid < EPB)               sSrc[tid]       = ei[ebase + tid];
  else if (tid < 2 * EPB)      sDst[tid - EPB] = ei[NE + ebase + tid - EPB];
  __syncthreads();

  const int le  = tid >> 4;             // local edge 0..15
  const int tg  = tid & 15;             // thread within edge
  const int c0  = tg * 4;               // channel group base
  const int src = sSrc[le];
  const int dst = sDst[le];

  float ep[4];
#pragma unroll
  for (int i = 0; i < 4; ++i) ep[i] = sbe[c0 + i];
#pragma unroll
  for (int j = 0; j < EDIM; ++j) {
    const float xej = sXE[le * EDIM + j];
#pragma unroll
    for (int i = 0; i < 4; ++i)
      ep[i] = fmaf(xej, sWe[j * DD + c0 + i], ep[i]);
  }

  const float4 kd4 = *(const float4*)&K[(size_t)dst * DD + c0];
  const float4 qs4 = *(const float4*)&Q[(size_t)src * DD + c0];
  const float4 vs4 = *(const float4*)&V[(size_t)src * DD + c0];
  const float kd[4] = {kd4.x, kd4.y, kd4.z, kd4.w};
  const float qs[4] = {qs4.x, qs4.y, qs4.z, qs4.w};
  const float vs[4] = {vs4.x, vs4.y, vs4.z, vs4.w};

#pragma unroll
  for (int i = 0; i < 4; ++i) {
    const float gate = kd[i] + qs[i] + 2.0f * ep[i];  // (k_i+ep)+(q_j+ep)
    const float sig  = 1.0f / (1.0f + __expf(-gate));
    const float msg  = sig * (vs[i] + ep[i]);
    atomicAdd(&AGG[(size_t)dst * DD + c0 + i], msg);
  }
}

// ---------------------------------------------------------------------------
// H = AGG + S (in place into AGG) and accumulate per-channel sum / sumsq for
// BatchNorm batch statistics. One block covers 64 rows; LDS reduction then
// 128 global atomics per block.
// ---------------------------------------------------------------------------
__global__ __launch_bounds__(256) void skip_stats_kernel(
    float* __restrict__ H,              // in: AGG, out: AGG + S
    const float* __restrict__ S,
    float* __restrict__ stats) {        // [0..63]=sum, [64..127]=sumsq
  __shared__ float ssum[DD];
  __shared__ float ssq[DD];
  const int c  = threadIdx.x & 63;
  const int r0 = threadIdx.x >> 6;      // 0..3
  if (threadIdx.x < DD) { ssum[threadIdx.x] = 0.0f; ssq[threadIdx.x] = 0.0f; }
  __syncthreads();

  float lsum = 0.0f, lsq = 0.0f;
  const int rowbase = blockIdx.x * 64;
  for (int r = r0; r < 64; r += 4) {
    const int row = rowbase + r;
    if (row < NN) {
      const size_t idx = (size_t)row * DD + c;
      const float h = H[idx] + S[idx];
      H[idx] = h;
      lsum += h;
      lsq  += h * h;
    }
  }
  atomicAdd(&ssum[c], lsum);
  atomicAdd(&ssq[c], lsq);
  __syncthreads();
  if (threadIdx.x < DD) {
    atomicAdd(&stats[threadIdx.x], ssum[threadIdx.x]);
    atomicAdd(&stats[DD + threadIdx.x], ssq[threadIdx.x]);
  }
}

// ---------------------------------------------------------------------------
// BN(training-mode batch stats) + ReLU, elementwise, float4-vectorized.
// ---------------------------------------------------------------------------
__global__ __launch_bounds__(256) void bn_relu_kernel(
    const float* __restrict__ H,
    const float* __restrict__ stats,
    const float* __restrict__ gamma,
    const float* __restrict__ beta,
    float* __restrict__ out) {
  const int q = blockIdx.x * 256 + threadIdx.x;   // index in float4 units
  if (q >= NN * DD / 4) return;
  const int base = q * 4;
  const int c0   = base & 63;
  const float invN = 1.0f / (float)NN;

  const float4 h4 = *(const float4*)&H[base];
  const float h[4] = {h4.x, h4.y, h4.z, h4.w};
  float y[4];
#pragma unroll
  for (int i = 0; i < 4; ++i) {
    const int c = c0 + i;
    const float mean = stats[c] * invN;
    const float var  = stats[DD + c] * invN - mean * mean;
    const float inv  = rsqrtf(var + EPS);
    y[i] = fmaxf((h[i] - mean) * inv * gamma[c] + beta[c], 0.0f);
  }
  float4 o4 = {y[0], y[1], y[2], y[3]};
  *(float4*)&out[base] = o4;
}

// ---------------------------------------------------------------------------
// Host orchestration.
// Input order (setup_inputs dict insertion order):
//   0:x  1:edge_index  2:x_edge
//   conv1 (3..12):  Wk bk Wq bq Wv bv We be Ws bs
//   conv2 (13..22): Wk bk Wq bq Wv bv We be Ws bs
//   23:bn1_g 24:bn1_b 25:bn2_g 26:bn2_b
// ---------------------------------------------------------------------------
extern "C" void kernel_launch(void* const* d_in, const int* in_sizes, int n_in,
                              void* d_out, int out_size, void* d_ws,
                              size_t ws_size, hipStream_t stream) {
  (void)in_sizes; (void)n_in; (void)out_size; (void)ws_size;

  const float* x  = (const float*)d_in[0];
  const int*   ei = (const int*)d_in[1];
  const float* xe = (const float*)d_in[2];

  const size_t NM = (size_t)NN * DD;
  float* Kb    = (float*)d_ws;
  float* Qb    = Kb + NM;
  float* Vb    = Qb + NM;
  float* Sb    = Vb + NM;
  float* AGG   = Sb + NM;
  float* H1    = AGG + NM;
  float* stats = H1 + NM;

  auto layer = [&](const float* xin, int p, const float* g, const float* b,
                   float* xout) {
    const float* Wk = (const float*)d_in[p + 0];
    const float* bk = (const float*)d_in[p + 1];
    const float* Wq = (const float*)d_in[p + 2];
    const float* bq = (const float*)d_in[p + 3];
    const float* Wv = (const float*)d_in[p + 4];
    const float* bv = (const float*)d_in[p + 5];
    const float* We = (const float*)d_in[p + 6];
    const float* be = (const float*)d_in[p + 7];
    const float* Ws = (const float*)d_in[p + 8];
    const float* bs = (const float*)d_in[p + 9];

    zero_kernel<<<(int)((NM + 255) / 256), 256, 0, stream>>>(AGG, stats,
                                                             (int)NM);

    dim3 gBlk(32, 16);
    gemm_node_wmma<<<NN / 16, gBlk, 0, stream>>>(xin, Wk, bk, Wq, bq, Wv, bv,
                                                 Ws, bs, Kb, Qb, Vb, Sb);

    edge_kernel<<<NE / EPB, 256, 0, stream>>>(ei, xe, We, be, Kb, Qb, Vb,
                                              AGG);

    skip_stats_kernel<<<(NN + 63) / 64, 256, 0, stream>>>(AGG, Sb, stats);

    bn_relu_kernel<<<(int)(NM / 4 + 255) / 256, 256, 0, stream>>>(AGG, stats,
                                                                  g, b, xout);
  };

  layer(x, 3, (const float*)d_in[23], (const float*)d_in[24], H1);
  layer(H1, 13, (const float*)d_in[25], (const float*)d_in[26],
        (float*)d_out);
}